// CustomGATConv_996432413191
// MI455X (gfx1250) — compile-verified
//
#include <hip/hip_runtime.h>
#include <hip/hip_bf16.h>

typedef __attribute__((ext_vector_type(2))) float v2f;
typedef __attribute__((ext_vector_type(8))) float v8f;

#define IN_C   128
#define OC     128   // OUT_C * HEADS
#define HEADS  4
#define HC     32
#define NEG_SLOPE 0.2f

// Native no-return fp32 atomic add, device scope (L2 atomic units).
// Tracked by STOREcnt; s_endpgm's implicit wait-idle covers completion.
__device__ __forceinline__ void gatom_add_f32(float* p, float v)
{
    asm volatile("global_atomic_add_f32 %0, %1, off scope:SCOPE_DEV"
                 :: "v"(p), "v"(v) : "memory");
}

// ---------------------------------------------------------------------------
// Kernel 1: x_proj[n, o] = sum_k x[n,k] * W[o,k]   via V_WMMA_F32_16X16X4_F32
// One wave computes a 16-node x 128-channel strip (8 C/D tiles, K-loop of 32).
// ---------------------------------------------------------------------------
__global__ __launch_bounds__(256) void gat_gemm_wmma(
    const float* __restrict__ x, const float* __restrict__ W,
    float* __restrict__ xproj, int N)
{
    const int lane = threadIdx.x & 31;
    const int wave = threadIdx.x >> 5;
    const int m    = lane & 15;
    const int kh   = lane >> 4;          // which K-half of the fragment (0/1)

    const int node_base = blockIdx.x * 128 + wave * 16;   // uniform per wave
    if (node_base >= N) return;          // wave-uniform: EXEC stays all-1s

    int arow = node_base + m;
    if (arow >= N) arow = N - 1;         // clamp (pad rows discarded at store)

    v8f acc[8] = {};                     // 8 tiles of 16x16 f32 accumulators

    const float* xr = x + (long)arow * IN_C + kh * 2;
#pragma unroll 4
    for (int k0 = 0; k0 < IN_C; k0 += 4) {
        v2f a = *(const v2f*)(xr + k0);
#pragma unroll
        for (int t = 0; t < 8; ++t) {
            const float* wr = W + (long)(t * 16 + m) * IN_C + k0 + kh * 2;
            v2f b = *(const v2f*)wr;
            acc[t] = __builtin_amdgcn_wmma_f32_16x16x4_f32(
                false, a, false, b, (short)0, acc[t], false, false);
        }
    }

    // D layout: VGPR v, lanes 0-15 -> row v, lanes 16-31 -> row v+8; col = m
#pragma unroll
    for (int t = 0; t < 8; ++t) {
#pragma unroll
        for (int v = 0; v < 8; ++v) {
            int r = node_base + v + 8 * kh;
            if (r < N) xproj[(long)r * OC + t * 16 + m] = acc[t][v];
        }
    }
}

// ---------------------------------------------------------------------------
// Kernel 2: per-(node,head) attention dots  al = <xproj, att_l>, ar likewise
// ---------------------------------------------------------------------------
__global__ void gat_att(const float* __restrict__ xproj,
                        const float* __restrict__ att_l,
                        const float* __restrict__ att_r,
                        float* __restrict__ al, float* __restrict__ ar, int NH)
{
    int i = blockIdx.x * blockDim.x + threadIdx.x;   // i = n*HEADS + h
    if (i >= NH) return;
    int h = i & (HEADS - 1);
    const float* xp = xproj + (long)(i >> 2) * OC + h * HC;
    const float* L  = att_l + h * HC;
    const float* R  = att_r + h * HC;
    float sl = 0.f, sr = 0.f;
#pragma unroll
    for (int c = 0; c < HC; ++c) { float v = xp[c]; sl += v * L[c]; sr += v * R[c]; }
    al[i] = sl; ar[i] = sr;
}

// ---------------------------------------------------------------------------
// Kernel 3: per-(edge,head)  alpha_exp = exp(leaky(al[row]+ar[col])),
//           denom[row,h] += alpha_exp   (native fp32 global atomics)
// ---------------------------------------------------------------------------
__global__ void gat_edge_alpha(const int* __restrict__ row,
                               const int* __restrict__ col,
                               const float* __restrict__ al,
                               const float* __restrict__ ar,
                               float* __restrict__ aex,
                               float* __restrict__ denom, long EH)
{
    long i = (long)blockIdx.x * blockDim.x + threadIdx.x;  // i = e*HEADS + h
    if (i >= EH) return;
    long e = i >> 2; int h = (int)(i & (HEADS - 1));
    int r = row[e], c = col[e];
    float a = al[r * HEADS + h] + ar[c * HEADS + h];
    a = (a >= 0.f) ? a : NEG_SLOPE * a;
    float ex = expf(a);
    aex[i] = ex;
    gatom_add_f32(&denom[r * HEADS + h], ex);
}

// ---------------------------------------------------------------------------
// Kernel 4: denom -> 1/(denom + eps)  in place (so scatter multiplies)
// ---------------------------------------------------------------------------
__global__ void gat_rdenom(float* __restrict__ denom, int NH)
{
    int i = blockIdx.x * blockDim.x + threadIdx.x;
    if (i < NH) denom[i] = 1.0f / (denom[i] + 1e-16f);
}

// ---------------------------------------------------------------------------
// Kernel 5: weighted scatter-add. One wave per edge; each lane owns 4
// consecutive channels:
//   - edge index is wave-uniform -> readfirstlane -> row/col via s_load
//   - x_proj[col] gather: one global_load_b128 per lane (512B/wave, coalesced)
//   - 4 native fp32 atomics per lane into out[row]
// All operands L2-resident (xproj 51MB + out 51MB << 192MB L2).
// ---------------------------------------------------------------------------
__global__ __launch_bounds__(256) void gat_scatter(
    const int* __restrict__ row, const int* __restrict__ col,
    const float* __restrict__ xproj, const float* __restrict__ aex,
    const float* __restrict__ rden, float* __restrict__ out, long E)
{
    // one wave (32 lanes) per edge; 8 waves per block
    int e = __builtin_amdgcn_readfirstlane(blockIdx.x * 8 + (threadIdx.x >> 5));
    if (e >= (int)E) return;             // wave-uniform exit
    const int lane = threadIdx.x & 31;
    const int ch   = lane * 4;           // 4 channels per lane
    const int h    = lane >> 3;          // head = ch / 32

    int r = row[e], c = col[e];          // scalar loads (e, ptrs uniform)
    float alpha = aex[(long)e * HEADS + h] * rden[r * HEADS + h];

    const float4 xj = *(const float4*)(xproj + (long)c * OC + ch);
    float* po = out + (long)r * OC + ch;
    gatom_add_f32(po + 0, xj.x * alpha);
    gatom_add_f32(po + 1, xj.y * alpha);
    gatom_add_f32(po + 2, xj.z * alpha);
    gatom_add_f32(po + 3, xj.w * alpha);
}

// ---------------------------------------------------------------------------
// Kernel 0: out = bias (broadcast), and a zero-fill for denom
// ---------------------------------------------------------------------------
__global__ void gat_init_out(float* __restrict__ out,
                             const float* __restrict__ bias, long n)
{
    long i = (long)blockIdx.x * blockDim.x + threadIdx.x;
    if (i < n) out[i] = bias[i & (OC - 1)];
}

__global__ void gat_zero(float* __restrict__ p, int n)
{
    int i = blockIdx.x * blockDim.x + threadIdx.x;
    if (i < n) p[i] = 0.0f;
}

// ---------------------------------------------------------------------------
extern "C" void kernel_launch(void* const* d_in, const int* in_sizes, int n_in,
                              void* d_out, int out_size, void* d_ws, size_t ws_size,
                              hipStream_t stream)
{
    const float* x     = (const float*)d_in[0];   // [N,128]
    const int*   ei    = (const int*)  d_in[1];   // [2,E]
    const float* W     = (const float*)d_in[2];   // [128,128]
    const float* att_l = (const float*)d_in[3];   // [1,4,32]
    const float* att_r = (const float*)d_in[4];   // [1,4,32]
    const float* bias  = (const float*)d_in[5];   // [128]

    const int  N = in_sizes[0] / IN_C;
    const long E = in_sizes[1] / 2;
    const int* row = ei;
    const int* col = ei + E;

    // workspace layout (floats): xproj | al | ar | denom | alpha_exp
    float* ws    = (float*)d_ws;
    float* xproj = ws;                               // N*128
    float* al    = xproj + (long)N * OC;             // N*4
    float* ar    = al    + (long)N * HEADS;          // N*4
    float* denom = ar    + (long)N * HEADS;          // N*4
    float* aex   = denom + (long)N * HEADS;          // E*4
    float* out   = (float*)d_out;

    const int NH = N * HEADS;
    const long EH = E * HEADS;

    gat_init_out<<<(unsigned)(((long)N * OC + 255) / 256), 256, 0, stream>>>(
        out, bias, (long)N * OC);
    gat_zero<<<(NH + 255) / 256, 256, 0, stream>>>(denom, NH);

    gat_gemm_wmma<<<(N + 127) / 128, 256, 0, stream>>>(x, W, xproj, N);

    gat_att<<<(NH + 255) / 256, 256, 0, stream>>>(xproj, att_l, att_r, al, ar, NH);

    gat_edge_alpha<<<(unsigned)((EH + 255) / 256), 256, 0, stream>>>(
        row, col, al, ar, aex, denom, EH);

    gat_rdenom<<<(NH + 255) / 256, 256, 0, stream>>>(denom, NH);

    // one wave per edge, 8 waves (256 threads) per block
    gat_scatter<<<(unsigned)((E + 7) / 8), 256, 0, stream>>>(
        row, col, xproj, aex, denom, out, E);
}